// AR_GAS_89189290869335
// MI455X (gfx1250) — compile-verified
//
#include <hip/hip_runtime.h>
#include <stdint.h>
#include <math.h>

// AR-GAS Student-t score-driven scan, K = 4M steps.
// Strictly serial nonlinear recurrence -> single wave32, dependent chain on
// VALU, async global->LDS DMA (gfx1250 ASYNCcnt path) for input staging.
// Per-step output capture via readfirstlane + v_writelane (no mask SGPRs, no
// spills), sqrt + coalesced stores off the critical path.

#define CHUNK 512                      // floats per LDS buffer (2 KB)
#define LOADS_PER_CHUNK (CHUNK / 32)   // 16 x global_load_async_to_lds_b32 (128B each)

__global__ __launch_bounds__(32)
void ar_gas_scan_kernel(const float* __restrict__ y,
                        const float* p_last_mu,  const float* p_last_s2,
                        const float* p_alpha_mu, const float* p_alpha_s,
                        const float* p_beta_mu,  const float* p_beta_s,
                        const float* p_omega_mu, const float* p_omega_s,
                        const float* p_nu,       const float* p_strength,
                        float* __restrict__ out, int n)
{
    __shared__ float buf[2][CHUNK];
    const int lane = (int)threadIdx.x;

    // ---- fold constants once (off the hot chain) ----
    const float strength = *p_strength;
    const float nu    = *p_nu;
    const float a_mu  = (*p_alpha_mu) * strength;
    const float a_s   = (*p_alpha_s)  * strength;
    const float b_mu  = *p_beta_mu;
    const float b_s   = *p_beta_s;
    const float w_mu  = *p_omega_mu;
    const float w_s   = *p_omega_s;
    const float inv_nu = 1.0f / nu;          // one exact divide, done once
    const float c1    = 1.0f + inv_nu;
    const float D_mu  = b_mu * a_mu * c1;    // coeff of r*p in mu update
    const float C_s   = b_s * (1.0f - a_s);  // coeff of s2 in s2 update
    const float D_s   = b_s * a_s * c1;      // coeff of r^2*p in s2 update

    float mu = *p_last_mu;
    float s2 = *p_last_s2;                   // "last_sigma" is a variance

    float* out_mu  = out;
    float* out_sig = out + n;

    // one serial GAS step; mu/s2 are wave-uniform (all lanes redundant)
    auto step = [&](float yl) {
        const float r  = yl - mu;
        const float r2 = r * r;
        const float tt = nu * s2;
        const float q  = fmaf(r, r, tt);                 // r^2 + nu*s2
        const float p  = tt * __builtin_amdgcn_rcpf(q);  // = t/q  (v_rcp_f32)
        const float u  = fmaf(b_mu, mu, w_mu);           // parallel
        const float v  = fmaf(C_s,  s2, w_s);            // parallel
        mu = fmaf(D_mu * r,  p, u);
        s2 = fmaf(D_s  * r2, p, v);
    };

    const int nchunk = n / CHUNK;
    const int n_main = nchunk * CHUNK;

    // per-lane LDS byte addresses (AS3 offset == low 32 bits of flat addr)
    const uint32_t lds0 = (uint32_t)(uintptr_t)(&buf[0][lane]);
    const uint32_t lds1 = (uint32_t)(uintptr_t)(&buf[1][lane]);

    // make the LDS object "written" + escaped, drain DS before async DMA
    buf[0][lane] = 0.0f;
    buf[1][lane] = 0.0f;
    asm volatile("s_wait_dscnt 0" ::: "memory");

    // prologue: kick chunk 0
    if (nchunk > 0) {
        const uint64_t g = (uint64_t)(uintptr_t)(y + lane);
        #pragma unroll
        for (int i = 0; i < LOADS_PER_CHUNK; ++i)
            asm volatile("global_load_async_to_lds_b32 %0, %1, off offset:%2"
                         :: "v"(lds0), "v"(g), "n"(i * 128) : "memory");
    }

    #pragma unroll 1
    for (int c = 0; c < nchunk; ++c) {
        const int  cur       = c & 1;
        const bool have_next = (c + 1) < nchunk;

        if (have_next) {  // overlap next chunk's DMA with this chunk's math
            const uint64_t g = (uint64_t)(uintptr_t)(y + (c + 1) * CHUNK + lane);
            const uint32_t l = cur ? lds0 : lds1;
            #pragma unroll
            for (int i = 0; i < LOADS_PER_CHUNK; ++i)
                asm volatile("global_load_async_to_lds_b32 %0, %1, off offset:%2"
                             :: "v"(l), "v"(g), "n"(i * 128) : "memory");
            // in-order completion: <=16 outstanding means chunk c has landed
            asm volatile("s_wait_asynccnt %0" :: "n"(LOADS_PER_CHUNK) : "memory");
        } else {
            asm volatile("s_wait_asynccnt 0" ::: "memory");
        }

        const float* bc = &buf[cur][0];
        #pragma unroll 1
        for (int t = 0; t < CHUNK / 32; ++t) {
            // pull the whole tile into registers up-front (pipelined ds loads,
            // keeps LDS latency off the serial chain)
            float yv[32];
            #pragma unroll
            for (int l = 0; l < 32; ++l) yv[l] = bc[t * 32 + l];

            int muKb = 0, s2Kb = 0;
            #pragma unroll
            for (int l = 0; l < 32; ++l) {
                step(yv[l]);
                // capture step l's (uniform) state into lane l of the output
                // registers: readfirstlane -> SGPR, v_writelane -> lane l.
                {
                    const int muS = __builtin_amdgcn_readfirstlane(__float_as_int(mu));
                    const int s2S = __builtin_amdgcn_readfirstlane(__float_as_int(s2));
                    asm("v_writelane_b32 %0, %1, %2" : "+v"(muKb) : "s"(muS), "n"(l));
                    asm("v_writelane_b32 %0, %1, %2" : "+v"(s2Kb) : "s"(s2S), "n"(l));
                }
            }
            const int gb = c * CHUNK + t * 32;
            out_mu[gb + lane]  = __int_as_float(muKb);
            out_sig[gb + lane] = sqrtf(__int_as_float(s2Kb));  // off-chain
        }
    }

    // tail: n not a multiple of CHUNK (K=4M divides exactly; kept for
    // generality). Per-lane coalesced load + shfl broadcast, no scalar loads.
    #pragma unroll 1
    for (int base = n_main; base < n; base += 32) {
        const int idx = base + lane;
        const int rem = n - base;
        const float ym = (idx < n) ? y[idx] : 0.0f;

        int muKb = 0, s2Kb = 0;
        #pragma unroll
        for (int l = 0; l < 32; ++l) {
            if (l < rem) {                       // wave-uniform condition
                step(__shfl(ym, l, 32));
                const int muS = __builtin_amdgcn_readfirstlane(__float_as_int(mu));
                const int s2S = __builtin_amdgcn_readfirstlane(__float_as_int(s2));
                asm("v_writelane_b32 %0, %1, %2" : "+v"(muKb) : "s"(muS), "n"(l));
                asm("v_writelane_b32 %0, %1, %2" : "+v"(s2Kb) : "s"(s2S), "n"(l));
            }
        }
        if (idx < n) {
            out_mu[idx]  = __int_as_float(muKb);
            out_sig[idx] = sqrtf(__int_as_float(s2Kb));
        }
    }
}

extern "C" void kernel_launch(void* const* d_in, const int* in_sizes, int n_in,
                              void* d_out, int out_size, void* d_ws, size_t ws_size,
                              hipStream_t stream) {
    // setup_inputs() order:
    // 0 deep_preds, 1 last_mu, 2 last_sigma, 3 alpha_mu, 4 alpha_sigma,
    // 5 beta_mu, 6 beta_sigma, 7 omega_mu, 8 omega_sigma, 9 nu, 10 norm_strength
    const float* y          = (const float*)d_in[0];
    const float* last_mu    = (const float*)d_in[1];
    const float* last_sigma = (const float*)d_in[2];
    const float* alpha_mu   = (const float*)d_in[3];
    const float* alpha_s    = (const float*)d_in[4];
    const float* beta_mu    = (const float*)d_in[5];
    const float* beta_s     = (const float*)d_in[6];
    const float* omega_mu   = (const float*)d_in[7];
    const float* omega_s    = (const float*)d_in[8];
    const float* nu         = (const float*)d_in[9];
    const float* strength   = (const float*)d_in[10];
    float* out = (float*)d_out;
    const int n = in_sizes[0];

    // strictly serial recurrence: one wave32 owns the whole chain
    ar_gas_scan_kernel<<<1, 32, 0, stream>>>(
        y, last_mu, last_sigma, alpha_mu, alpha_s, beta_mu, beta_s,
        omega_mu, omega_s, nu, strength, out, n);
}